// GNNML3_64991445123464
// MI455X (gfx1250) — compile-verified
//
#include <hip/hip_runtime.h>

typedef __attribute__((ext_vector_type(2))) float v2f;
typedef __attribute__((ext_vector_type(8))) float v8f;

static constexpr int N_NODES = 50000;
static constexpr int N_EDGES = 1600000;
static constexpr int KFREQ   = 5;
static constexpr int N_GRAPH = 250;
static constexpr int C_CONV  = 64;
static constexpr int C_FEAT  = 66;   // 64 conv + 2 gate
static constexpr int KPAD2   = 332;  // 5*66 = 330 padded to multiple of 4

__global__ void zero_f32(float* __restrict__ p, long n) {
  long i = (long)blockIdx.x * blockDim.x + threadIdx.x;
  if (i < n) p[i] = 0.0f;
}

// Copy stacked weight [ktot,64] into zero-padded [kpad,64]
__global__ void pad_w(const float* __restrict__ W, float* __restrict__ B, int ktot, int kpad) {
  int i = blockIdx.x * blockDim.x + threadIdx.x;
  if (i >= kpad * C_CONV) return;
  int k = i >> 6;
  B[i] = (k < ktot) ? W[i] : 0.0f;
}

// Layer 1 scatter: x is [N,1], Hcat row stride 8 (5 used)
__global__ void scatter_l1(const float* __restrict__ x, const int* __restrict__ src,
                           const int* __restrict__ dst, const float* __restrict__ ea,
                           float* __restrict__ H) {
  int e = blockIdx.x * blockDim.x + threadIdx.x;
  if (e >= N_EDGES) return;
  float xv = x[src[e]];
  long base = (long)dst[e] * 8;
#pragma unroll
  for (int k = 0; k < KFREQ; ++k)
    atomicAdd(&H[base + k], ea[e * KFREQ + k] * xv);
}

// Layers 2/3 scatter: x is [N,66], Hcat row stride 332, layout [node][kf*66 + c]
__global__ void scatter_l23(const float* __restrict__ x, const int* __restrict__ src,
                            const int* __restrict__ dst, const float* __restrict__ ea,
                            float* __restrict__ H) {
  long t = (long)blockIdx.x * blockDim.x + threadIdx.x;
  if (t >= (long)N_EDGES * C_FEAT) return;
  int e = (int)(t / C_FEAT);
  int c = (int)(t - (long)e * C_FEAT);
  float xv = x[(long)src[e] * C_FEAT + c];
  long base = (long)dst[e] * KPAD2 + c;
#pragma unroll
  for (int k = 0; k < KFREQ; ++k)
    atomicAdd(&H[base + k * C_FEAT], ea[e * KFREQ + k] * xv);
}

// One wave per 16-node tile. conv = relu(A @ B + bias) via V_WMMA_F32_16X16X4_F32
// (full fp32 — matches reference numerics). Also computes the 2-channel gate.
// A: Hcat[16 x KPAD] (zero-padded K). B: padded stacked W [KPAD x 64].
// ISA layouts (wave32):
//   A 16x4 f32 : lane<16 -> {A[lm][k], A[lm][k+1]}, lane>=16 -> {A[lm][k+2], A[lm][k+3]}
//   B 4x16 f32 : lane<16 -> {B[k][lm], B[k+1][lm]}, lane>=16 -> {B[k+2][lm], B[k+3][lm]}
//   C/D 16x16  : VGPR j -> row j (lanes<16) / row j+8 (lanes>=16), col = lane&15
template <int NIN, int KPAD>
__global__ __launch_bounds__(32) void combine_wmma(
    const float* __restrict__ H, const float* __restrict__ B,
    const float* __restrict__ bias, const float* __restrict__ xin,
    const float* __restrict__ f11W, const float* __restrict__ f11b,
    const float* __restrict__ f12W, const float* __restrict__ f12b,
    float* __restrict__ xout) {
  const int lane = threadIdx.x;
  const int hi = lane >> 4;
  const int lm = lane & 15;
  const int m0 = blockIdx.x * 16;

  v8f c0 = {}, c1 = {}, c2 = {}, c3 = {};
  const float* arow = H + (long)(m0 + lm) * KPAD + 2 * hi;
  for (int k = 0; k < KPAD; k += 4) {
    v2f a;
    a.x = arow[k];
    a.y = arow[k + 1];
    const float* b0p = B + (k + 2 * hi) * C_CONV + lm;
    const float* b1p = b0p + C_CONV;
    v2f b;
    b.x = b0p[0];  b.y = b1p[0];
    c0 = __builtin_amdgcn_wmma_f32_16x16x4_f32(false, a, false, b, (short)0, c0, false, false);
    b.x = b0p[16]; b.y = b1p[16];
    c1 = __builtin_amdgcn_wmma_f32_16x16x4_f32(false, a, false, b, (short)0, c1, false, false);
    b.x = b0p[32]; b.y = b1p[32];
    c2 = __builtin_amdgcn_wmma_f32_16x16x4_f32(false, a, false, b, (short)0, c2, false, false);
    b.x = b0p[48]; b.y = b1p[48];
    c3 = __builtin_amdgcn_wmma_f32_16x16x4_f32(false, a, false, b, (short)0, c3, false, false);
  }

  const float bb0 = bias[lm], bb1 = bias[lm + 16], bb2 = bias[lm + 32], bb3 = bias[lm + 48];
#pragma unroll
  for (int j = 0; j < 8; ++j) {
    float* orow = xout + (long)(m0 + j + 8 * hi) * C_FEAT;
    orow[lm]      = fmaxf(c0[j] + bb0, 0.0f);
    orow[lm + 16] = fmaxf(c1[j] + bb1, 0.0f);
    orow[lm + 32] = fmaxf(c2[j] + bb2, 0.0f);
    orow[lm + 48] = fmaxf(c3[j] + bb3, 0.0f);
  }

  // gate: lane -> (node = m0+lm, out channel = hi)
  const float* xr = xin + (long)(m0 + lm) * NIN;
  float g1 = f11b[hi], g2 = f12b[hi];
#pragma unroll
  for (int c = 0; c < NIN; ++c) {
    float xv = xr[c];
    g1 = fmaf(xv, f11W[c * 2 + hi], g1);
    g2 = fmaf(xv, f12W[c * 2 + hi], g2);
  }
  xout[(long)(m0 + lm) * C_FEAT + C_CONV + hi] = fmaxf(g1, 0.0f) * fmaxf(g2, 0.0f);
}

__global__ void pool_sum(const float* __restrict__ x, const int* __restrict__ batch,
                         float* __restrict__ psum, float* __restrict__ pcnt) {
  long t = (long)blockIdx.x * blockDim.x + threadIdx.x;
  if (t >= (long)N_NODES * C_FEAT) return;
  int n = (int)(t / C_FEAT);
  int c = (int)(t - (long)n * C_FEAT);
  int g = batch[n];
  atomicAdd(&psum[g * C_FEAT + c], x[t]);
  if (c == 0) atomicAdd(&pcnt[g], 1.0f);
}

__global__ void head_mlp(const float* __restrict__ psum, const float* __restrict__ pcnt,
                         const float* __restrict__ fc1W, const float* __restrict__ fc1b,
                         const float* __restrict__ fc2W, const float* __restrict__ fc2b,
                         float* __restrict__ out) {
  int g = blockIdx.x * blockDim.x + threadIdx.x;
  if (g >= N_GRAPH) return;
  float inv = 1.0f / fmaxf(pcnt[g], 1.0f);
  float h[32];
#pragma unroll
  for (int j = 0; j < 32; ++j) h[j] = fc1b[j];
  for (int c = 0; c < C_FEAT; ++c) {
    float v = psum[g * C_FEAT + c] * inv;
#pragma unroll
    for (int j = 0; j < 32; ++j) h[j] = fmaf(v, fc1W[c * 32 + j], h[j]);
  }
  float o = fc2b[0];
#pragma unroll
  for (int j = 0; j < 32; ++j) o = fmaf(fmaxf(h[j], 0.0f), fc2W[j], o);
  out[g] = o;
}

static inline unsigned cdiv(long a, long b) { return (unsigned)((a + b - 1) / b); }

extern "C" void kernel_launch(void* const* d_in, const int* in_sizes, int n_in,
                              void* d_out, int out_size, void* d_ws, size_t ws_size,
                              hipStream_t stream) {
  (void)in_sizes; (void)n_in; (void)out_size; (void)ws_size;
  const float* x0   = (const float*)d_in[0];
  const int*   ei   = (const int*)d_in[1];
  const int*   src  = ei;
  const int*   dst  = ei + N_EDGES;
  const float* ea   = (const float*)d_in[2];
  const int*   batch = (const int*)d_in[3];
  const float* cW[3]   = {(const float*)d_in[4],  (const float*)d_in[10], (const float*)d_in[16]};
  const float* cb[3]   = {(const float*)d_in[5],  (const float*)d_in[11], (const float*)d_in[17]};
  const float* f11W[3] = {(const float*)d_in[6],  (const float*)d_in[12], (const float*)d_in[18]};
  const float* f11b[3] = {(const float*)d_in[7],  (const float*)d_in[13], (const float*)d_in[19]};
  const float* f12W[3] = {(const float*)d_in[8],  (const float*)d_in[14], (const float*)d_in[20]};
  const float* f12b[3] = {(const float*)d_in[9],  (const float*)d_in[15], (const float*)d_in[21]};
  const float* fc1W = (const float*)d_in[22];
  const float* fc1b = (const float*)d_in[23];
  const float* fc2W = (const float*)d_in[24];
  const float* fc2b = (const float*)d_in[25];

  char* ws = (char*)d_ws;
  float* Hcat = (float*)ws;  ws += (size_t)N_NODES * KPAD2 * sizeof(float);   // 66.4 MB
  float* xa   = (float*)ws;  ws += (size_t)N_NODES * C_FEAT * sizeof(float);  // 13.2 MB
  float* xb   = (float*)ws;  ws += (size_t)N_NODES * C_FEAT * sizeof(float);  // 13.2 MB
  float* Bpad = (float*)ws;  ws += (size_t)KPAD2 * C_CONV * sizeof(float);    // 85 KB
  float* psum = (float*)ws;  ws += (size_t)N_GRAPH * C_FEAT * sizeof(float);  // 66 KB
  float* pcnt = (float*)ws;                                                    // 1 KB (contig after psum)

  // ---- layer 1 (nin = 1, K = 5 -> pad 8) ----
  pad_w<<<cdiv(8 * C_CONV, 256), 256, 0, stream>>>(cW[0], Bpad, 5, 8);
  zero_f32<<<cdiv((long)N_NODES * 8, 256), 256, 0, stream>>>(Hcat, (long)N_NODES * 8);
  scatter_l1<<<cdiv(N_EDGES, 256), 256, 0, stream>>>(x0, src, dst, ea, Hcat);
  combine_wmma<1, 8><<<N_NODES / 16, 32, 0, stream>>>(
      Hcat, Bpad, cb[0], x0, f11W[0], f11b[0], f12W[0], f12b[0], xa);

  // ---- layer 2 (nin = 66, K = 330 -> pad 332) ----
  pad_w<<<cdiv(KPAD2 * C_CONV, 256), 256, 0, stream>>>(cW[1], Bpad, 330, KPAD2);
  zero_f32<<<cdiv((long)N_NODES * KPAD2, 256), 256, 0, stream>>>(Hcat, (long)N_NODES * KPAD2);
  scatter_l23<<<cdiv((long)N_EDGES * C_FEAT, 256), 256, 0, stream>>>(xa, src, dst, ea, Hcat);
  combine_wmma<C_FEAT, KPAD2><<<N_NODES / 16, 32, 0, stream>>>(
      Hcat, Bpad, cb[1], xa, f11W[1], f11b[1], f12W[1], f12b[1], xb);

  // ---- layer 3 ----
  pad_w<<<cdiv(KPAD2 * C_CONV, 256), 256, 0, stream>>>(cW[2], Bpad, 330, KPAD2);
  zero_f32<<<cdiv((long)N_NODES * KPAD2, 256), 256, 0, stream>>>(Hcat, (long)N_NODES * KPAD2);
  scatter_l23<<<cdiv((long)N_EDGES * C_FEAT, 256), 256, 0, stream>>>(xb, src, dst, ea, Hcat);
  combine_wmma<C_FEAT, KPAD2><<<N_NODES / 16, 32, 0, stream>>>(
      Hcat, Bpad, cb[2], xb, f11W[2], f11b[2], f12W[2], f12b[2], xa);

  // ---- mean pool + head ----
  zero_f32<<<cdiv((long)N_GRAPH * (C_FEAT + 1), 256), 256, 0, stream>>>(
      psum, (long)N_GRAPH * (C_FEAT + 1));  // psum and pcnt are contiguous
  pool_sum<<<cdiv((long)N_NODES * C_FEAT, 256), 256, 0, stream>>>(xa, batch, psum, pcnt);
  head_mlp<<<cdiv(N_GRAPH, 128), 128, 0, stream>>>(psum, pcnt, fc1W, fc1b, fc2W, fc2b,
                                                   (float*)d_out);
}